// SAT_GSL_49959059587770
// MI455X (gfx1250) — compile-verified
//
#include <hip/hip_runtime.h>
#include <math.h>

// CDNA5 / gfx1250. wave32. WMMA f32 16x16x4, software-pipelined operand loads.
typedef __attribute__((ext_vector_type(2))) float v2f;
typedef __attribute__((ext_vector_type(8))) float v8f;

#define DIMD 256          // feature dim D
#define WAVES_PER_BLOCK 8 // 256 threads = 8 wave32

// log(2) - softplus(-p); softplus(x) = max(x,0) + log1p(exp(-|x|)) (stable, matches jax)
__device__ __forceinline__ float jsd_pos(float p) {
    float x  = -p;
    float sp = fmaxf(x, 0.0f) + log1pf(expf(-fabsf(x)));
    return 0.69314718f - sp; // np.float32(log(2))
}

__device__ __forceinline__ v2f ld2(const float* p)   { return *(const v2f*)p; }
__device__ __forceinline__ v2f ld2nt(const float* p) { return __builtin_nontemporal_load((const v2f*)p); }

__global__ void gsl_init(float* __restrict__ accum) {
    if (threadIdx.x < 2) accum[threadIdx.x] = 0.0f;
}

// One wave handles 16 nodes. D = A*B + C with:
//   A (16x4 f32): lane m = lane&15 holds z[row0+m][k + 2*(lane>>4) .. +1] (ISA 16x4 A layout)
//   B (4x16 f32): column n = gathered g-row for node row0+n; same float2 addressing per lane
// diag(D) after K=256 accumulation = the 16 gathered dot products.
__global__ void __launch_bounds__(256)
gsl_tile_kernel(const int* __restrict__ b1, const int* __restrict__ b2,
                const float* __restrict__ z1, const float* __restrict__ z2,
                const float* __restrict__ g1, const float* __restrict__ g2,
                float* __restrict__ accum, int n)
{
    const int lane = threadIdx.x & 31;
    const int wave = threadIdx.x >> 5;
    const int tile = blockIdx.x * WAVES_PER_BLOCK + wave;
    const int row0 = tile * 16;
    if (row0 >= n) return;                 // wave-uniform: EXEC stays all-ones for WMMA

    const int m    = lane & 15;
    const int hi   = lane >> 4;            // selects K pair {0,1} vs {2,3}
    const int row  = row0 + m;
    const int rowc = (row < n) ? row : (n - 1);   // clamp loads for a partial tail tile

    const int i1 = b1[rowc];
    const int i2 = b2[rowc];
    const float* z1r = z1 + (size_t)rowc * DIMD;  // streamed once -> non-temporal
    const float* z2r = z2 + (size_t)rowc * DIMD;  // streamed once -> non-temporal
    const float* gA  = g1 + (size_t)i1 * DIMD;    // d11 = z1 . g1[b1]   (reused, RT)
    const float* gB  = g2 + (size_t)i1 * DIMD;    // d12 = z1 . g2[b1]
    const float* gC  = g2 + (size_t)i2 * DIMD;    // d22 = z2 . g2[b2]
    const float* gD  = g1 + (size_t)i2 * DIMD;    // d21 = z2 . g1[b2]

    // Warm the latency-exposed indirect g-row gathers (global_prefetch_b8).
    __builtin_prefetch(gA, 0, 0);
    __builtin_prefetch(gB, 0, 0);
    __builtin_prefetch(gC, 0, 0);
    __builtin_prefetch(gD, 0, 0);

    v8f a11 = {0.f,0.f,0.f,0.f,0.f,0.f,0.f,0.f};
    v8f a12 = a11, a22 = a11, a21 = a11;

    const int base = 2 * hi;

    // ---- software pipeline: stage k's operands live while k+1's loads are in flight ----
    v2f cz1 = ld2nt(z1r + base);
    v2f cz2 = ld2nt(z2r + base);
    v2f cA  = ld2(gA + base);
    v2f cB  = ld2(gB + base);
    v2f cC  = ld2(gC + base);
    v2f cD  = ld2(gD + base);

    #pragma unroll 4
    for (int k = 0; k < DIMD - 4; k += 4) {
        const int nidx = k + 4 + base;     // even -> 8B-aligned float2 loads
        // issue next-step loads first: keeps ~6 loads outstanding instead of wait(0) per WMMA
        v2f nz1 = ld2nt(z1r + nidx);
        v2f nz2 = ld2nt(z2r + nidx);
        v2f nA  = ld2(gA + nidx);
        v2f nB  = ld2(gB + nidx);
        v2f nC  = ld2(gC + nidx);
        v2f nD  = ld2(gD + nidx);

        a11 = __builtin_amdgcn_wmma_f32_16x16x4_f32(false, cz1, false, cA, (short)0, a11, false, false);
        a12 = __builtin_amdgcn_wmma_f32_16x16x4_f32(false, cz1, false, cB, (short)0, a12, false, false);
        a22 = __builtin_amdgcn_wmma_f32_16x16x4_f32(false, cz2, false, cC, (short)0, a22, false, false);
        a21 = __builtin_amdgcn_wmma_f32_16x16x4_f32(false, cz2, false, cD, (short)0, a21, false, false);

        cz1 = nz1; cz2 = nz2; cA = nA; cB = nB; cC = nC; cD = nD;
    }
    // epilogue: last k-step
    a11 = __builtin_amdgcn_wmma_f32_16x16x4_f32(false, cz1, false, cA, (short)0, a11, false, false);
    a12 = __builtin_amdgcn_wmma_f32_16x16x4_f32(false, cz1, false, cB, (short)0, a12, false, false);
    a22 = __builtin_amdgcn_wmma_f32_16x16x4_f32(false, cz2, false, cC, (short)0, a22, false, false);
    a21 = __builtin_amdgcn_wmma_f32_16x16x4_f32(false, cz2, false, cD, (short)0, a21, false, false);

    // Diagonal extraction (C/D layout: VGPR r = rows M=r (lanes 0-15) and M=r+8 (lanes 16-31)):
    //   diag(i), i<8  -> acc[i]   on lane i
    //   diag(i), i>=8 -> acc[i-8] on lane i+16   (i.e. lanes 24..31 hold acc[lane-24])
    float d11 = 0.f, d12 = 0.f, d22 = 0.f, d21 = 0.f;
    #pragma unroll
    for (int r = 0; r < 8; ++r) {
        const bool sel = (lane == r) || (lane == r + 24);
        d11 = sel ? a11[r] : d11;
        d12 = sel ? a12[r] : d12;
        d22 = sel ? a22[r] : d22;
        d21 = sel ? a21[r] : d21;
    }

    const int node   = (lane < 8) ? (row0 + lane) : (row0 + lane - 16);
    const bool diagl = (lane < 8) || (lane >= 24);
    const bool active = diagl && (node < n);

    const float t1 = jsd_pos(d11) - jsd_pos(d12);
    const float t2 = jsd_pos(d22) - jsd_pos(d21);
    float s1 = active ? t1 * t1 : 0.0f;
    float s2 = active ? t2 * t2 : 0.0f;

    #pragma unroll
    for (int off = 16; off > 0; off >>= 1) {
        s1 += __shfl_xor(s1, off, 32);
        s2 += __shfl_xor(s2, off, 32);
    }
    if (lane == 0) {
        atomicAdd(&accum[0], s1);
        atomicAdd(&accum[1], s2);
    }
}

__global__ void gsl_final(const float* __restrict__ accum, float* __restrict__ out) {
    if (threadIdx.x == 0) out[0] = sqrtf(accum[0]) + sqrtf(accum[1]);
}

extern "C" void kernel_launch(void* const* d_in, const int* in_sizes, int n_in,
                              void* d_out, int out_size, void* d_ws, size_t ws_size,
                              hipStream_t stream) {
    const int*   b1 = (const int*)d_in[0];
    const int*   b2 = (const int*)d_in[1];
    const float* z1 = (const float*)d_in[2];
    const float* z2 = (const float*)d_in[3];
    const float* g1 = (const float*)d_in[4];
    const float* g2 = (const float*)d_in[5];
    float* out   = (float*)d_out;
    float* accum = (float*)d_ws;

    const int n      = in_sizes[0];                 // 65536 nodes
    const int tiles  = (n + 15) / 16;
    const int blocks = (tiles + WAVES_PER_BLOCK - 1) / WAVES_PER_BLOCK;

    hipLaunchKernelGGL(gsl_init,  dim3(1), dim3(32), 0, stream, accum);
    hipLaunchKernelGGL(gsl_tile_kernel, dim3(blocks), dim3(WAVES_PER_BLOCK * 32), 0, stream,
                       b1, b2, z1, z2, g1, g2, accum, n);
    hipLaunchKernelGGL(gsl_final, dim3(1), dim3(32), 0, stream, accum, out);
}